// MessagePassingNetwork_90658169684165
// MI455X (gfx1250) — compile-verified
//
#include <hip/hip_runtime.h>
#include <math.h>

#define N_NODES 100000
#define N_EDGES 3200000
#define DH 20
#define N_TILES (N_NODES / 16)   // 6250 exactly

typedef __attribute__((ext_vector_type(2))) float v2f;
typedef __attribute__((ext_vector_type(8))) float v8f;

// epilogue modes
#define EP_NONE 0
#define EP_RELU 1
#define EP_SIGMOID 2

// ---------------------------------------------------------------------------
// Node-level GEMM via V_WMMA_F32_16X16X4_F32 (full f32 precision).
// out[N_NODES, NCOLS] = ep( (A * rowscale) @ W + bias ), rowscale = 1/deg (mean)
// One wave per 16-row tile; K in {4,20}; NCOLS in {20,3}. Branchless B loads.
// ---------------------------------------------------------------------------
template <int K, int NCOLS, int MODE, bool USE_CNT>
__global__ __launch_bounds__(256) void node_gemm_wmma(
    const float* __restrict__ A,     // [N_NODES, K]
    const float* __restrict__ W,     // [K, NCOLS]
    const float* __restrict__ bias,  // [NCOLS]
    const float* __restrict__ deg,   // [N_NODES] (only if USE_CNT)
    float* __restrict__ out)         // [N_NODES, NCOLS]
{
  const int wave = threadIdx.x >> 5;
  const int tile = blockIdx.x * 8 + wave;
  if (tile >= N_TILES) return;  // wave-uniform: EXEC stays all-1s for WMMA

  const int lane = threadIdx.x & 31;
  const int half = lane >> 4;
  const int lq   = lane & 15;
  const int rowBase = tile * 16;
  const int arow = rowBase + lq;

  float ascale = 1.0f;
  if (USE_CNT) ascale = 1.0f / fmaxf(deg[arow], 1.0f);

  // branchless column clamps + masks (keeps EXEC all-1s, no saveexec blocks)
  const int   cb0 = (lq < NCOLS) ? lq : (NCOLS - 1);
  const float m0  = (lq < NCOLS) ? 1.0f : 0.0f;
  const int   c1  = 16 + lq;
  const int   cb1 = (c1 < NCOLS) ? c1 : (NCOLS - 1);
  const float m1  = (c1 < NCOLS) ? 1.0f : 0.0f;

  v8f acc0 = {};
  v8f acc1 = {};

#pragma unroll
  for (int k0 = 0; k0 < K; k0 += 4) {
    const int ka = k0 + 2 * half;
    v2f a;
    a.x = A[arow * K + ka + 0] * ascale;
    a.y = A[arow * K + ka + 1] * ascale;

    v2f b0;
    b0.x = W[(ka + 0) * NCOLS + cb0] * m0;
    b0.y = W[(ka + 1) * NCOLS + cb0] * m0;
    acc0 = __builtin_amdgcn_wmma_f32_16x16x4_f32(false, a, false, b0,
                                                 (short)0, acc0, false, false);
    if (NCOLS > 16) {
      v2f b1;
      b1.x = W[(ka + 0) * NCOLS + cb1] * m1;
      b1.y = W[(ka + 1) * NCOLS + cb1] * m1;
      acc1 = __builtin_amdgcn_wmma_f32_16x16x4_f32(false, a, false, b1,
                                                   (short)0, acc1, false, false);
    }
  }

#pragma unroll
  for (int v = 0; v < 8; ++v) {
    const int row = rowBase + v + 8 * half;
    float zr = 1.0f;
    if (USE_CNT) zr = (deg[row] > 0.0f) ? 1.0f : 0.0f;  // where(cnt>0, ., 0)
    if (lq < NCOLS) {
      float r = (acc0[v] + bias[lq]) * zr;
      if (MODE == EP_RELU) r = fmaxf(r, 0.0f);
      if (MODE == EP_SIGMOID) r = 1.0f / (1.0f + __expf(-r));
      out[row * NCOLS + lq] = r;
    }
    if (NCOLS > 16 && c1 < NCOLS) {
      float r = (acc1[v] + bias[c1]) * zr;
      if (MODE == EP_RELU) r = fmaxf(r, 0.0f);
      if (MODE == EP_SIGMOID) r = 1.0f / (1.0f + __expf(-r));
      out[row * NCOLS + c1] = r;
    }
  }
}

// ---------------------------------------------------------------------------
// CSR build (once per launch): histogram -> single-block scan -> slot scatter
// ---------------------------------------------------------------------------
__global__ __launch_bounds__(256) void hist_kernel(
    const int* __restrict__ dst, int* __restrict__ cnt)
{
  const int e = blockIdx.x * blockDim.x + threadIdx.x;
  if (e >= N_EDGES) return;
  atomicAdd(&cnt[dst[e]], 1);
}

__global__ __launch_bounds__(1024) void scan_kernel(
    const int* __restrict__ cnt, int* __restrict__ rowptr,
    int* __restrict__ cursor, float* __restrict__ deg)
{
  __shared__ int tmp[1024];
  __shared__ int carry;
  const int tid = threadIdx.x;
  if (tid == 0) carry = 0;
  __syncthreads();
  for (int base = 0; base < N_NODES; base += 1024) {
    const int i = base + tid;
    const int v = (i < N_NODES) ? cnt[i] : 0;
    tmp[tid] = v;
    __syncthreads();
#pragma unroll
    for (int off = 1; off < 1024; off <<= 1) {
      const int t = (tid >= off) ? tmp[tid - off] : 0;
      __syncthreads();
      tmp[tid] += t;
      __syncthreads();
    }
    const int incl = tmp[tid];
    const int base_off = carry;  // stable: last write was before prior barrier
    if (i < N_NODES) {
      const int excl = base_off + incl - v;
      rowptr[i] = excl;
      cursor[i] = excl;
      deg[i] = (float)v;
    }
    __syncthreads();
    if (tid == 1023) carry = base_off + incl;
    __syncthreads();
  }
  if (tid == 0) rowptr[N_NODES] = N_EDGES;
}

__global__ __launch_bounds__(256) void scatter_kernel(
    const int* __restrict__ src, const int* __restrict__ dst,
    const float* __restrict__ attr, int* __restrict__ cursor,
    int* __restrict__ es, float* __restrict__ ea)
{
  const int e = blockIdx.x * blockDim.x + threadIdx.x;
  if (e >= N_EDGES) return;
  const int d = dst[e];
  const int slot = atomicAdd(&cursor[d], 1);
  es[slot] = src[e];
  ea[slot] = attr[e];
}

// ---------------------------------------------------------------------------
// Per-node aggregation over CSR edge range (no atomics):
// ACC[n] = sum_{e in bucket(n)} relu(U[es[e]] + ea[e] * colsum(W1))
// U (8 MB) is fully L2-resident on MI455X (192 MB L2) -> gathers hit L2.
// ---------------------------------------------------------------------------
__global__ __launch_bounds__(256) void aggregate_kernel(
    const float* __restrict__ U,       // [N_NODES, 20]
    const int* __restrict__ rowptr,    // [N_NODES+1]
    const int* __restrict__ es,        // [N_EDGES] sorted-by-dst src ids
    const float* __restrict__ ea,      // [N_EDGES] sorted-by-dst attrs
    const float* __restrict__ w1s,     // [20] column sums of W1
    float* __restrict__ acc)           // [N_NODES, 20]
{
  __shared__ __align__(16) float sw[DH];
  if (threadIdx.x < DH) sw[threadIdx.x] = w1s[threadIdx.x];
  __syncthreads();

  const int n = blockIdx.x * blockDim.x + threadIdx.x;
  if (n >= N_NODES) return;
  const int beg = rowptr[n];
  const int end = rowptr[n + 1];

  float4 a0 = {0, 0, 0, 0}, a1 = {0, 0, 0, 0}, a2 = {0, 0, 0, 0};
  float4 a3 = {0, 0, 0, 0}, a4 = {0, 0, 0, 0};
  const float4 w0 = ((const float4*)sw)[0];
  const float4 w1 = ((const float4*)sw)[1];
  const float4 w2 = ((const float4*)sw)[2];
  const float4 w3 = ((const float4*)sw)[3];
  const float4 w4 = ((const float4*)sw)[4];

  for (int i = beg; i < end; ++i) {
    const int s = es[i];
    const float a = ea[i];
    const float4* up = (const float4*)(U + s * DH);
    const float4 u0 = up[0], u1 = up[1], u2 = up[2], u3 = up[3], u4 = up[4];
    a0.x += fmaxf(u0.x + a * w0.x, 0.0f); a0.y += fmaxf(u0.y + a * w0.y, 0.0f);
    a0.z += fmaxf(u0.z + a * w0.z, 0.0f); a0.w += fmaxf(u0.w + a * w0.w, 0.0f);
    a1.x += fmaxf(u1.x + a * w1.x, 0.0f); a1.y += fmaxf(u1.y + a * w1.y, 0.0f);
    a1.z += fmaxf(u1.z + a * w1.z, 0.0f); a1.w += fmaxf(u1.w + a * w1.w, 0.0f);
    a2.x += fmaxf(u2.x + a * w2.x, 0.0f); a2.y += fmaxf(u2.y + a * w2.y, 0.0f);
    a2.z += fmaxf(u2.z + a * w2.z, 0.0f); a2.w += fmaxf(u2.w + a * w2.w, 0.0f);
    a3.x += fmaxf(u3.x + a * w3.x, 0.0f); a3.y += fmaxf(u3.y + a * w3.y, 0.0f);
    a3.z += fmaxf(u3.z + a * w3.z, 0.0f); a3.w += fmaxf(u3.w + a * w3.w, 0.0f);
    a4.x += fmaxf(u4.x + a * w4.x, 0.0f); a4.y += fmaxf(u4.y + a * w4.y, 0.0f);
    a4.z += fmaxf(u4.z + a * w4.z, 0.0f); a4.w += fmaxf(u4.w + a * w4.w, 0.0f);
  }
  float4* ap = (float4*)(acc + n * DH);
  ap[0] = a0; ap[1] = a1; ap[2] = a2; ap[3] = a3; ap[4] = a4;
}

__global__ void colsum_kernel(const float* __restrict__ W, int K,
                              float* __restrict__ out)
{
  const int j = threadIdx.x;
  if (j < DH) {
    float s = 0.0f;
    for (int k = 0; k < K; ++k) s += W[k * DH + j];
    out[j] = s;
  }
}

extern "C" void kernel_launch(void* const* d_in, const int* in_sizes, int n_in,
                              void* d_out, int out_size, void* d_ws,
                              size_t ws_size, hipStream_t stream)
{
  const float* x    = (const float*)d_in[0];
  const int*   ei   = (const int*)d_in[1];
  const float* attr = (const float*)d_in[2];
  const float* W1a = (const float*)d_in[3],  *b1a = (const float*)d_in[4];
  const float* W2a = (const float*)d_in[5],  *b2a = (const float*)d_in[6];
  const float* W1b = (const float*)d_in[7],  *b1b = (const float*)d_in[8];
  const float* W2b = (const float*)d_in[9],  *b2b = (const float*)d_in[10];
  const float* W1c = (const float*)d_in[11], *b1c = (const float*)d_in[12];
  const float* W2c = (const float*)d_in[13], *b2c = (const float*)d_in[14];
  const float* Wc  = (const float*)d_in[15], *bc  = (const float*)d_in[16];
  const int* src = ei;
  const int* dst = ei + N_EDGES;
  float* out = (float*)d_out;

  // Workspace carve-out (256B aligned)
  char* ws = (char*)d_ws;
  size_t off = 0;
  auto carve = [&](size_t bytes) {
    void* p = ws + off;
    off += (bytes + 255) & ~(size_t)255;
    return p;
  };
  float* U      = (float*)carve((size_t)N_NODES * DH * 4);
  float* ACC    = (float*)carve((size_t)N_NODES * DH * 4);
  float* H      = (float*)carve((size_t)N_NODES * DH * 4);
  float* deg    = (float*)carve((size_t)N_NODES * 4);
  int*   cnt    = (int*)  carve((size_t)N_NODES * 4);
  int*   rowptr = (int*)  carve((size_t)(N_NODES + 1) * 4);
  int*   cursor = (int*)  carve((size_t)N_NODES * 4);
  int*   es     = (int*)  carve((size_t)N_EDGES * 4);
  float* ea     = (float*)carve((size_t)N_EDGES * 4);
  float* w1sa   = (float*)carve(DH * 4);
  float* w1sb   = (float*)carve(DH * 4);
  float* w1sc   = (float*)carve(DH * 4);
  (void)in_sizes; (void)n_in; (void)out_size; (void)ws_size;

  const dim3 eb(256), eg((N_EDGES + 255) / 256);
  const dim3 gb(256), gg((N_TILES + 7) / 8);
  const dim3 nb(256), ng((N_NODES + 255) / 256);

  // ---- one-time per launch: dst-CSR + colsums (shared by all 3 layers) ----
  hipMemsetAsync(cnt, 0, (size_t)N_NODES * 4, stream);
  hist_kernel<<<eg, eb, 0, stream>>>(dst, cnt);
  scan_kernel<<<1, 1024, 0, stream>>>(cnt, rowptr, cursor, deg);
  scatter_kernel<<<eg, eb, 0, stream>>>(src, dst, attr, cursor, es, ea);
  colsum_kernel<<<1, 32, 0, stream>>>(W1a, 4,  w1sa);
  colsum_kernel<<<1, 32, 0, stream>>>(W1b, DH, w1sb);
  colsum_kernel<<<1, 32, 0, stream>>>(W1c, DH, w1sc);

  // ---- layer a ----
  node_gemm_wmma<4, DH, EP_NONE, false><<<gg, gb, 0, stream>>>(x, W1a, b1a, nullptr, U);
  aggregate_kernel<<<ng, nb, 0, stream>>>(U, rowptr, es, ea, w1sa, ACC);
  node_gemm_wmma<DH, DH, EP_RELU, true><<<gg, gb, 0, stream>>>(ACC, W2a, b2a, deg, H);

  // ---- layer b ----
  node_gemm_wmma<DH, DH, EP_NONE, false><<<gg, gb, 0, stream>>>(H, W1b, b1b, nullptr, U);
  aggregate_kernel<<<ng, nb, 0, stream>>>(U, rowptr, es, ea, w1sb, ACC);
  node_gemm_wmma<DH, DH, EP_RELU, true><<<gg, gb, 0, stream>>>(ACC, W2b, b2b, deg, H);

  // ---- layer c (no inter-layer relu on its output) ----
  node_gemm_wmma<DH, DH, EP_NONE, false><<<gg, gb, 0, stream>>>(H, W1c, b1c, nullptr, U);
  aggregate_kernel<<<ng, nb, 0, stream>>>(U, rowptr, es, ea, w1sc, ACC);
  node_gemm_wmma<DH, DH, EP_NONE, true><<<gg, gb, 0, stream>>>(ACC, W2c, b2c, deg, H);

  // ---- head: out = sigmoid(h @ Wc + bc), [N,3] ----
  node_gemm_wmma<DH, 3, EP_SIGMOID, false><<<gg, gb, 0, stream>>>(H, Wc, bc, nullptr, out);
}